// ConvolutionalLayer_28157805593468
// MI455X (gfx1250) — compile-verified
//
#include <hip/hip_runtime.h>

typedef __attribute__((ext_vector_type(2))) float v2f;
typedef __attribute__((ext_vector_type(8))) float v8f;

#define DIM 128
#define NTILES 8           // 128 / 16 column tiles per wave
#define WAVES_PER_BLOCK 8  // 256 threads / wave32

// ---------------------------------------------------------------------------
// Kernel 1: out[n, d] = bias[d]   (so the scatter-adds produce segsum + bias)
// ---------------------------------------------------------------------------
__global__ __launch_bounds__(256) void gcn_init_out(float* __restrict__ out,
                                                    const float* __restrict__ bias,
                                                    int total) {
    int i = blockIdx.x * blockDim.x + threadIdx.x;
    if (i < total) out[i] = bias[i & (DIM - 1)];
}

// ---------------------------------------------------------------------------
// Kernel 2: h = x @ W  via V_WMMA_F32_16X16X4_F32.
// One wave computes a 16x128 row panel; W staged in LDS (64 KB).
// ---------------------------------------------------------------------------
__global__ __launch_bounds__(256) void gcn_gemm_wmma(const float* __restrict__ x,
                                                     const float* __restrict__ w,
                                                     float* __restrict__ h,
                                                     int nrows) {
    __shared__ float wlds[DIM * DIM];  // 64 KB

    // Cooperative stage of W into LDS (4096 float4 / 256 threads = 16 each)
    {
        const float4* wsrc = reinterpret_cast<const float4*>(w);
        float4* wdst = reinterpret_cast<float4*>(wlds);
        #pragma unroll
        for (int i = threadIdx.x; i < DIM * DIM / 4; i += 256) wdst[i] = wsrc[i];
    }
    __syncthreads();

    const int wave = threadIdx.x >> 5;
    const int lane = threadIdx.x & 31;
    const int half = lane >> 4;   // 0: lanes 0-15, 1: lanes 16-31
    const int l15  = lane & 15;

    const int mtiles = (nrows + 15) >> 4;
    const int mtile  = blockIdx.x * WAVES_PER_BLOCK + wave;
    if (mtile >= mtiles) return;  // whole-wave exit: EXEC all-1s at WMMAs

    const int row  = mtile * 16 + l15;
    const int lrow = row < nrows ? row : nrows - 1;  // clamp (loads only)

    v8f acc[NTILES];
    const v8f vzero = {0.f, 0.f, 0.f, 0.f, 0.f, 0.f, 0.f, 0.f};
    #pragma unroll
    for (int n = 0; n < NTILES; ++n) acc[n] = vzero;

    const int koff = half * 2;  // A/B 32-bit fragment layout: K split across halves

    for (int kb = 0; kb < DIM; kb += 4) {
        // A fragment: lane holds A[M=l15][kb+koff], A[M=l15][kb+koff+1]
        const float2 af =
            *reinterpret_cast<const float2*>(x + (size_t)lrow * DIM + kb + koff);
        v2f a;
        a.x = af.x;
        a.y = af.y;

        #pragma unroll
        for (int n = 0; n < NTILES; ++n) {
            const int col = n * 16 + l15;
            v2f b;
            b.x = wlds[(kb + koff    ) * DIM + col];
            b.y = wlds[(kb + koff + 1) * DIM + col];
            // D = A(16x4) * B(4x16) + C
            acc[n] = __builtin_amdgcn_wmma_f32_16x16x4_f32(
                /*neg_a=*/false, a, /*neg_b=*/false, b,
                /*c_mod=*/(short)0, acc[n],
                /*reuse_a=*/false, /*reuse_b=*/false);
        }
    }

    // D layout: VGPR r -> M = r + half*8, N = l15 (per 16x16 f32 C/D layout)
    #pragma unroll
    for (int n = 0; n < NTILES; ++n) {
        #pragma unroll
        for (int r = 0; r < 8; ++r) {
            const int m = mtile * 16 + half * 8 + r;
            if (m < nrows) h[(size_t)m * DIM + n * 16 + l15] = acc[n][r];
        }
    }
}

// ---------------------------------------------------------------------------
// Kernel 3: COO scatter: out[dst] += w_e * h[src].  One lane per (edge, 4-chunk).
// h and out both fit in the 192 MB L2 -> gather + atomics are L2-resident.
// ---------------------------------------------------------------------------
__global__ __launch_bounds__(256) void gcn_scatter(const float* __restrict__ h,
                                                   const int* __restrict__ esrc,
                                                   const int* __restrict__ edst,
                                                   const float* __restrict__ ew,
                                                   float* out, int E) {
    const long long t = (long long)blockIdx.x * blockDim.x + threadIdx.x;
    const int e = (int)(t >> 5);
    if (e >= E) return;
    const int c = (int)(t & 31) * 4;

    const int s = esrc[e];
    const int d = edst[e];
    const float wt = ew[e];

    const float4 v = *reinterpret_cast<const float4*>(h + (size_t)s * DIM + c);
    float* o = out + (size_t)d * DIM + c;

    __hip_atomic_fetch_add(o + 0, v.x * wt, __ATOMIC_RELAXED, __HIP_MEMORY_SCOPE_AGENT);
    __hip_atomic_fetch_add(o + 1, v.y * wt, __ATOMIC_RELAXED, __HIP_MEMORY_SCOPE_AGENT);
    __hip_atomic_fetch_add(o + 2, v.z * wt, __ATOMIC_RELAXED, __HIP_MEMORY_SCOPE_AGENT);
    __hip_atomic_fetch_add(o + 3, v.w * wt, __ATOMIC_RELAXED, __HIP_MEMORY_SCOPE_AGENT);
}

// ---------------------------------------------------------------------------
extern "C" void kernel_launch(void* const* d_in, const int* in_sizes, int n_in,
                              void* d_out, int out_size, void* d_ws, size_t ws_size,
                              hipStream_t stream) {
    const float* x    = (const float*)d_in[0];
    const int*   esrc = (const int*)d_in[1];
    const int*   edst = (const int*)d_in[2];
    const float* ew   = (const float*)d_in[3];
    const float* w    = (const float*)d_in[4];
    const float* bias = (const float*)d_in[5];

    const int N = in_sizes[0] / DIM;   // nodes
    const int E = in_sizes[1];         // edges

    float* out = (float*)d_out;
    float* h   = (float*)d_ws;         // N*DIM*4 bytes = 51.2 MB scratch

    // 1) out = bias broadcast
    {
        const int total = N * DIM;
        gcn_init_out<<<(total + 255) / 256, 256, 0, stream>>>(out, bias, total);
    }

    // 2) h = x @ W  (WMMA f32)
    {
        const int mtiles = (N + 15) >> 4;
        const int blocks = (mtiles + WAVES_PER_BLOCK - 1) / WAVES_PER_BLOCK;
        gcn_gemm_wmma<<<blocks, 256, 0, stream>>>(x, w, h, N);
    }

    // 3) out[dst] += w_e * h[src]
    {
        const long long threads = (long long)E * 32;
        const int blocks = (int)((threads + 255) / 256);
        gcn_scatter<<<blocks, 256, 0, stream>>>(h, esrc, edst, ew, out, E);
    }
}